// STGCN_5171140624834
// MI455X (gfx1250) — compile-verified
//
#include <hip/hip_runtime.h>

#define TT 12          // timesteps
#define CH 64          // hidden channels
#define COUT 32        // output channels
#define LDW 68         // padded LDS row stride (floats) -> conflict-free

typedef float v2f __attribute__((ext_vector_type(2)));
typedef float v8f __attribute__((ext_vector_type(8)));

// ---------------------------------------------------------------------------
// degree / edge-norm precompute
// ---------------------------------------------------------------------------
__global__ void zero_f32(float* __restrict__ p, int n) {
    int i = blockIdx.x * blockDim.x + threadIdx.x;
    if (i < n) p[i] = 0.0f;
}

__global__ void deg_accum(const int* __restrict__ dst, const float* __restrict__ ew,
                          float* __restrict__ deg, int nE) {
    int e = blockIdx.x * blockDim.x + threadIdx.x;
    if (e < nE) unsafeAtomicAdd(&deg[dst[e]], ew[e]);
}

__global__ void edge_norm(const int* __restrict__ src, const int* __restrict__ dst,
                          const float* __restrict__ ew, const float* __restrict__ deg,
                          float* __restrict__ nrm, int nE) {
    int e = blockIdx.x * blockDim.x + threadIdx.x;
    if (e >= nE) return;
    float ds = deg[src[e]] + 1.0f;   // +1 = self-loop weight
    float dd = deg[dst[e]] + 1.0f;
    nrm[e] = ew[e] * rsqrtf(ds) * rsqrtf(dd);
}

// ---------------------------------------------------------------------------
// fused temporal conv1d(k=3, pad=1) + InstanceNorm1d(affine) + ReLU
// in : [T, N, CIN]   out : [T, N, 64]
// one thread per (node, out_channel); all T=12 timesteps in registers
// ---------------------------------------------------------------------------
template <int CIN>
__global__ void tconv_in_relu(const float* __restrict__ x,
                              const float* __restrict__ cw,   // [64, CIN, 3]
                              const float* __restrict__ cb,   // [64]
                              const float* __restrict__ gamma,
                              const float* __restrict__ beta,
                              float* __restrict__ out, int nN) {
    int tid = blockIdx.x * blockDim.x + threadIdx.x;
    if (tid >= nN * CH) return;
    int oc = tid & (CH - 1);
    int n  = tid >> 6;

    float acc[TT];
    float b0 = cb[oc];
#pragma unroll
    for (int t = 0; t < TT; t++) acc[t] = b0;

    for (int ic = 0; ic < CIN; ic++) {
        float xv[TT];
#pragma unroll
        for (int t = 0; t < TT; t++)
            xv[t] = x[((size_t)t * nN + n) * CIN + ic];
        const float* wp = cw + ((size_t)oc * CIN + ic) * 3;
        float w0 = wp[0], w1 = wp[1], w2 = wp[2];
#pragma unroll
        for (int t = 0; t < TT; t++) {
            float s = xv[t] * w1;
            if (t > 0)      s += xv[t - 1] * w0;
            if (t < TT - 1) s += xv[t + 1] * w2;
            acc[t] += s;
        }
    }
    // InstanceNorm over T
    float mu = 0.f;
#pragma unroll
    for (int t = 0; t < TT; t++) mu += acc[t];
    mu *= (1.0f / TT);
    float var = 0.f;
#pragma unroll
    for (int t = 0; t < TT; t++) { float d = acc[t] - mu; var += d * d; }
    var *= (1.0f / TT);
    float rs = rsqrtf(var + 1e-5f);
    float g = gamma[oc], be = beta[oc];
#pragma unroll
    for (int t = 0; t < TT; t++) {
        float v = (acc[t] - mu) * rs * g + be;
        v = fmaxf(v, 0.0f);
        out[((size_t)t * nN + n) * CH + oc] = v;
    }
}

// ---------------------------------------------------------------------------
// WMMA GEMM: xw[t,n,:] = h[t,n,:] @ W (64x64), plus fused self-loop init:
// agg[t,n,:] = xw[t,n,:] / (deg[n]+1) + gb[:]
// W staged transposed in LDS (Wt[col*LDW + k]) so each B frag is one
// ds_load_b64.  Each wave owns TWO 16-node M-tiles: per k-step it issues
// 2 A-loads + 4 B-loads and fires 8 independent WMMAs.
// ---------------------------------------------------------------------------
__global__ void gcn_matmul_selfinit(const float* __restrict__ h,   // [T,N,64]
                                    const float* __restrict__ W,   // [64,64]
                                    const float* __restrict__ gb,  // [64]
                                    const float* __restrict__ deg, // [N]
                                    float* __restrict__ xw,        // [T,N,64]
                                    float* __restrict__ agg,       // [T,N,64]
                                    int nN) {
    __shared__ float Wt[CH * LDW];                 // 64 cols x 64 k, padded
    for (int i = threadIdx.x; i < CH * CH; i += blockDim.x) {
        int r = i >> 6, c = i & 63;                // W[r*64 + c]
        Wt[c * LDW + r] = W[i];
    }
    __syncthreads();

    int wave = threadIdx.x >> 5;
    int lane = threadIdx.x & 31;
    int nTiles = (nN + 15) >> 4;
    int nPairs = (nTiles + 1) >> 1;
    int tp = blockIdx.x * (blockDim.x >> 5) + wave;
    if (tp >= nPairs) return;                      // wave-uniform
    int t = blockIdx.y;
    int m0 = tp * 32;

    const float* hp = h + (size_t)t * nN * CH;
    float* xwp = xw + (size_t)t * nN * CH;
    float* agp = agg + (size_t)t * nN * CH;

    int khalf = (lane >> 4) * 2;                   // 0 or 2
    int ncol = lane & 15;
    int r0 = m0 + ncol;
    int r1 = r0 + 16;
    if (r0 >= nN) r0 = nN - 1;                     // clamp (N % 16 == 0 anyway)
    if (r1 >= nN) r1 = nN - 1;

    v8f acc0[4] = {};
    v8f acc1[4] = {};
#pragma unroll
    for (int kt = 0; kt < 16; kt++) {
        int kb = kt * 4 + khalf;
        v2f a0 = *(const v2f*)&hp[(size_t)r0 * CH + kb];
        v2f a1 = *(const v2f*)&hp[(size_t)r1 * CH + kb];
#pragma unroll
        for (int nt = 0; nt < 4; nt++) {
            v2f b = *(const v2f*)&Wt[(nt * 16 + ncol) * LDW + kb];
            acc0[nt] = __builtin_amdgcn_wmma_f32_16x16x4_f32(
                false, a0, false, b, (short)0, acc0[nt], false, false);
            acc1[nt] = __builtin_amdgcn_wmma_f32_16x16x4_f32(
                false, a1, false, b, (short)0, acc1[nt], false, false);
        }
    }
    // epilogue: C/D layout -> VGPR r: lanes0-15 = row r, lanes16-31 = row r+8
#pragma unroll
    for (int nt = 0; nt < 4; nt++) {
        int ch = nt * 16 + ncol;
        float bias = gb[ch];
#pragma unroll
        for (int r = 0; r < 8; r++) {
            int node0 = m0 + r + ((lane >> 4) << 3);
            int node1 = node0 + 16;
            if (node0 < nN) {
                float v = acc0[nt][r];
                float selfn = __builtin_amdgcn_rcpf(deg[node0] + 1.0f);
                xwp[(size_t)node0 * CH + ch] = v;
                agp[(size_t)node0 * CH + ch] = v * selfn + bias;
            }
            if (node1 < nN) {
                float v = acc1[nt][r];
                float selfn = __builtin_amdgcn_rcpf(deg[node1] + 1.0f);
                xwp[(size_t)node1 * CH + ch] = v;
                agp[(size_t)node1 * CH + ch] = v * selfn + bias;
            }
        }
    }
}

// ---------------------------------------------------------------------------
// edge scatter: agg[t,dst,:] += norm[e] * xw[t,src,:]
// 16 lanes x float4 per edge -> coalesced 64-float row read; native f32 atomics
// ---------------------------------------------------------------------------
__global__ void gcn_scatter(const float* __restrict__ xw, const float* __restrict__ nrm,
                            const int* __restrict__ src, const int* __restrict__ dst,
                            float* __restrict__ agg, int nN, int nE) {
    int tid = blockIdx.x * blockDim.x + threadIdx.x;   // over nE*16
    if (tid >= nE * 16) return;
    int cg = tid & 15;       // channel group of 4
    int e  = tid >> 4;
    int t  = blockIdx.y;
    int s = src[e], d = dst[e];
    float w = nrm[e];
    const float4* xv = (const float4*)(xw + ((size_t)t * nN + s) * CH) + cg;
    float* ap = agg + ((size_t)t * nN + d) * CH + cg * 4;
    __builtin_prefetch(ap, 1, 1);        // pull dst RMW line toward cache
    float4 v = *xv;
    unsafeAtomicAdd(ap + 0, v.x * w);
    unsafeAtomicAdd(ap + 1, v.y * w);
    unsafeAtomicAdd(ap + 2, v.z * w);
    unsafeAtomicAdd(ap + 3, v.w * w);
}

// ---------------------------------------------------------------------------
// hbar[n,c] = mean_t h[t,n,c]   (mean commutes with the final linear layer)
// ---------------------------------------------------------------------------
__global__ void temporal_mean(const float* __restrict__ h, float* __restrict__ hbar, int nN) {
    int tid = blockIdx.x * blockDim.x + threadIdx.x;
    if (tid >= nN * CH) return;
    float s = 0.f;
#pragma unroll
    for (int t = 0; t < TT; t++) s += h[(size_t)t * nN * CH + tid];
    hbar[tid] = s * (1.0f / TT);
}

// ---------------------------------------------------------------------------
// output WMMA GEMM: out[n,:] = hbar[n,:] @ out_w (64x32) + out_b
// same LDS-staged, two-M-tile structure as the GCN GEMM
// ---------------------------------------------------------------------------
__global__ void out_matmul(const float* __restrict__ hbar, const float* __restrict__ W, // [64,32]
                           const float* __restrict__ ob, float* __restrict__ out, int nN) {
    __shared__ float Wt[COUT * LDW];               // 32 cols x 64 k, padded
    for (int i = threadIdx.x; i < CH * COUT; i += blockDim.x) {
        int r = i >> 5, c = i & 31;                // W[r*32 + c]
        Wt[c * LDW + r] = W[i];
    }
    __syncthreads();

    int wave = threadIdx.x >> 5;
    int lane = threadIdx.x & 31;
    int nTiles = (nN + 15) >> 4;
    int nPairs = (nTiles + 1) >> 1;
    int tp = blockIdx.x * (blockDim.x >> 5) + wave;
    if (tp >= nPairs) return;
    int m0 = tp * 32;

    int khalf = (lane >> 4) * 2;
    int ncol = lane & 15;
    int r0 = m0 + ncol;
    int r1 = r0 + 16;
    if (r0 >= nN) r0 = nN - 1;
    if (r1 >= nN) r1 = nN - 1;

    v8f acc0[2] = {};
    v8f acc1[2] = {};
#pragma unroll
    for (int kt = 0; kt < 16; kt++) {
        int kb = kt * 4 + khalf;
        v2f a0 = *(const v2f*)&hbar[(size_t)r0 * CH + kb];
        v2f a1 = *(const v2f*)&hbar[(size_t)r1 * CH + kb];
#pragma unroll
        for (int nt = 0; nt < 2; nt++) {
            v2f b = *(const v2f*)&Wt[(nt * 16 + ncol) * LDW + kb];
            acc0[nt] = __builtin_amdgcn_wmma_f32_16x16x4_f32(
                false, a0, false, b, (short)0, acc0[nt], false, false);
            acc1[nt] = __builtin_amdgcn_wmma_f32_16x16x4_f32(
                false, a1, false, b, (short)0, acc1[nt], false, false);
        }
    }
#pragma unroll
    for (int nt = 0; nt < 2; nt++) {
        int ch = nt * 16 + ncol;
        float bias = ob[ch];
#pragma unroll
        for (int r = 0; r < 8; r++) {
            int node0 = m0 + r + ((lane >> 4) << 3);
            int node1 = node0 + 16;
            if (node0 < nN) out[(size_t)node0 * COUT + ch] = acc0[nt][r] + bias;
            if (node1 < nN) out[(size_t)node1 * COUT + ch] = acc1[nt][r] + bias;
        }
    }
}

// ---------------------------------------------------------------------------
extern "C" void kernel_launch(void* const* d_in, const int* in_sizes, int n_in,
                              void* d_out, int out_size, void* d_ws, size_t ws_size,
                              hipStream_t stream) {
    const float* x   = (const float*)d_in[0];
    const int*   ei  = (const int*)d_in[1];
    const float* ew  = (const float*)d_in[2];
    const float* cw1 = (const float*)d_in[3];
    const float* cb1 = (const float*)d_in[4];
    const float* g1  = (const float*)d_in[5];
    const float* be1 = (const float*)d_in[6];
    const float* gw1 = (const float*)d_in[7];
    const float* gb1 = (const float*)d_in[8];
    const float* cw2 = (const float*)d_in[9];
    const float* cb2 = (const float*)d_in[10];
    const float* g2  = (const float*)d_in[11];
    const float* be2 = (const float*)d_in[12];
    const float* gw2 = (const float*)d_in[13];
    const float* gb2 = (const float*)d_in[14];
    const float* ow  = (const float*)d_in[15];
    const float* ob  = (const float*)d_in[16];
    (void)n_in; (void)out_size; (void)ws_size;

    const int N = in_sizes[0] / (TT * 32);
    const int E = in_sizes[2];
    const int* src = ei;
    const int* dst = ei + E;

    // workspace carve-up
    char* ws = (char*)d_ws;
    size_t off = 0;
    auto carve = [&](size_t bytes) -> float* {
        float* p = (float*)(ws + off);
        off += (bytes + 255) & ~(size_t)255;
        return p;
    };
    const size_t hbytes = (size_t)TT * N * CH * sizeof(float);
    float* deg  = carve((size_t)N * sizeof(float));
    float* nrm  = carve((size_t)E * sizeof(float));
    float* bufA = carve(hbytes);   // relu output
    float* bufB = carve(hbytes);   // xw
    float* bufC = carve(hbytes);   // agg / next layer input
    float* hbar = carve((size_t)N * CH * sizeof(float));

    // degree + edge norm (recomputed each call: deg is accumulated via atomics)
    zero_f32<<<(N + 255) / 256, 256, 0, stream>>>(deg, N);
    deg_accum<<<(E + 255) / 256, 256, 0, stream>>>(dst, ew, deg, E);
    edge_norm<<<(E + 255) / 256, 256, 0, stream>>>(src, dst, ew, deg, nrm, E);

    const int nTiles = (N + 15) / 16;
    const int nPairs = (nTiles + 1) / 2;
    dim3 mmGrid((nPairs + 3) / 4, TT);
    dim3 scGrid((E * 16 + 255) / 256, TT);
    const int convBlocks = (N * CH + 255) / 256;

    // ----- layer 1 -----
    tconv_in_relu<32><<<convBlocks, 256, 0, stream>>>(x, cw1, cb1, g1, be1, bufA, N);
    gcn_matmul_selfinit<<<mmGrid, 128, 0, stream>>>(bufA, gw1, gb1, deg, bufB, bufC, N);
    gcn_scatter<<<scGrid, 256, 0, stream>>>(bufB, nrm, src, dst, bufC, N, E);

    // ----- layer 2 -----
    tconv_in_relu<64><<<convBlocks, 256, 0, stream>>>(bufC, cw2, cb2, g2, be2, bufA, N);
    gcn_matmul_selfinit<<<mmGrid, 128, 0, stream>>>(bufA, gw2, gb2, deg, bufB, bufC, N);
    gcn_scatter<<<scGrid, 256, 0, stream>>>(bufB, nrm, src, dst, bufC, N, E);

    // ----- output head -----
    temporal_mean<<<convBlocks, 256, 0, stream>>>(bufC, hbar, N);
    out_matmul<<<(nPairs + 3) / 4, 128, 0, stream>>>(hbar, ow, ob, (float*)d_out, N);
}